// MSA_21766894256177
// MI455X (gfx1250) — compile-verified
//
#include <hip/hip_runtime.h>
#include <hip/hip_bf16.h>

// B=4, N=2048, D=768, H=12, HD=64 ; scale = 1/8
#define Bk 4
#define Nk 2048
#define Dk 768
#define Hk 12

typedef __attribute__((ext_vector_type(2))) float v2f;
typedef __attribute__((ext_vector_type(8))) float v8f;

__device__ __forceinline__ v8f wmma_f32_16x16x4(v2f a, v2f b, v8f c) {
  // (neg_a, A, neg_b, B, c_mod, C, reuse_a, reuse_b)
  return __builtin_amdgcn_wmma_f32_16x16x4_f32(false, a, false, b, (short)0, c,
                                               false, false);
}

// ---------------------------------------------------------------------------
// Kernel 1: QKV projection.  qkv[b,h,n,e] = sum_d z[b,n,d]*Wqkv[h,e,d] + bqkv
// One wave computes a 32(n) x 64(e) macro-tile (2x4 register blocking):
// per K-step of 4: 2 A-frag + 4 B-frag b64 loads -> 8 WMMAs.
// ---------------------------------------------------------------------------
__global__ __launch_bounds__(128) void qkv_gemm_kernel(
    const float* __restrict__ z, const float* __restrict__ Wqkv,
    const float* __restrict__ bqkv, float* __restrict__ qkv) {
  const int wave = threadIdx.x >> 5;
  const int lane = threadIdx.x & 31;
  const int m = lane & 15;
  const int half = lane >> 4;

  int tile = blockIdx.x * 4 + wave;
  const int EB = 3;   // e blocks of 64 (192/64)
  const int NB = 64;  // n blocks of 32 (2048/32)
  int eb = tile % EB; tile /= EB;
  int nb = tile % NB; tile /= NB;
  int h = tile % Hk;
  int b = tile / Hk;

  // A fragments: rows nb*32 + {m, 16+m}; K at k + half*2 (+0,+1)
  const float* za0 = z + ((size_t)(b * Nk + nb * 32 + m)) * Dk + half * 2;
  const float* za1 = za0 + (size_t)16 * Dk;
  // B fragments: cols eb*64 + t*16 + m (Wqkv rows), K contiguous in d
  const float* wb = Wqkv + ((size_t)(h * 192 + eb * 64 + m)) * Dk + half * 2;

  v8f acc[2][4];
#pragma unroll
  for (int i = 0; i < 2; ++i)
#pragma unroll
    for (int t = 0; t < 4; ++t) acc[i][t] = (v8f){};

  for (int k0 = 0; k0 < Dk; k0 += 64) {
    // warm next K-chunk (emits global_prefetch_b8)
    __builtin_prefetch(za0 + k0 + 64, 0, 1);
    __builtin_prefetch(za1 + k0 + 64, 0, 1);
    __builtin_prefetch(wb + k0 + 64, 0, 1);
#pragma unroll
    for (int kk = 0; kk < 64; kk += 4) {
      const int k = k0 + kk;
      v2f a0 = *(const v2f*)(za0 + k);
      v2f a1 = *(const v2f*)(za1 + k);
      v2f b0 = *(const v2f*)(wb + k);
      v2f b1 = *(const v2f*)(wb + (size_t)16 * Dk + k);
      v2f b2 = *(const v2f*)(wb + (size_t)32 * Dk + k);
      v2f b3 = *(const v2f*)(wb + (size_t)48 * Dk + k);
      acc[0][0] = wmma_f32_16x16x4(a0, b0, acc[0][0]);
      acc[0][1] = wmma_f32_16x16x4(a0, b1, acc[0][1]);
      acc[0][2] = wmma_f32_16x16x4(a0, b2, acc[0][2]);
      acc[0][3] = wmma_f32_16x16x4(a0, b3, acc[0][3]);
      acc[1][0] = wmma_f32_16x16x4(a1, b0, acc[1][0]);
      acc[1][1] = wmma_f32_16x16x4(a1, b1, acc[1][1]);
      acc[1][2] = wmma_f32_16x16x4(a1, b2, acc[1][2]);
      acc[1][3] = wmma_f32_16x16x4(a1, b3, acc[1][3]);
    }
  }

  float bias[4];
#pragma unroll
  for (int t = 0; t < 4; ++t) bias[t] = bqkv[h * 192 + eb * 64 + t * 16 + m];

  float* outp =
      qkv + ((size_t)((b * Hk + h) * Nk + nb * 32)) * 192 + eb * 64 + m;
#pragma unroll
  for (int i = 0; i < 2; ++i)
#pragma unroll
    for (int r = 0; r < 8; ++r) {
      const size_t row = (size_t)(i * 16 + r + 8 * half) * 192;
#pragma unroll
      for (int t = 0; t < 4; ++t)
        outp[row + t * 16] = acc[i][t][r] + bias[t];
    }
}

// ---------------------------------------------------------------------------
// Kernel 2: flash attention, one wave per (b,h, 16-query tile).
// qkv layout per (b,h): row n has [q(0..63) | k(64..127) | v(128..191)].
// ---------------------------------------------------------------------------
__global__ __launch_bounds__(32) void attn_kernel(const float* __restrict__ qkv,
                                                  float* __restrict__ aout) {
  __shared__ float pt[16 * 17];  // padded P tile for C->A layout transpose
  const int lane = threadIdx.x & 31;
  const int m = lane & 15;
  const int half = lane >> 4;

  const int qt = blockIdx.x & 127;
  const int bh = blockIdx.x >> 7;  // b*H + h
  const int h = bh % Hk;
  const int b = bh / Hk;

  const float* base = qkv + (size_t)bh * Nk * 192;

  // Preload all q fragments for this 16x64 query tile (16 K-chunks of 4).
  v2f qa[16];
  const float* qp = base + (size_t)(qt * 16 + m) * 192 + half * 2;
#pragma unroll
  for (int kk = 0; kk < 16; ++kk) qa[kk] = *(const v2f*)(qp + kk * 4);

  v8f acc0 = {}, acc1 = {}, acc2 = {}, acc3 = {};
  float mrow[8], lrow[8];
#pragma unroll
  for (int r = 0; r < 8; ++r) { mrow[r] = -1e30f; lrow[r] = 0.0f; }
  const float scale = 0.125f;  // 1/sqrt(64)

  for (int kt = 0; kt < Nk / 16; ++kt) {
    // ---- S = q * k^T (16x16), K-dim = 64 -> 16 WMMAs
    const float* kp = base + (size_t)(kt * 16 + m) * 192 + 64 + half * 2;
    v8f s = {};
#pragma unroll
    for (int kk = 0; kk < 16; ++kk) {
      v2f bf = *(const v2f*)(kp + kk * 4);
      s = wmma_f32_16x16x4(qa[kk], bf, s);
    }

    // ---- online softmax; row r+8*half lives in vgpr r across 16 lanes
    float alpha[8];
#pragma unroll
    for (int r = 0; r < 8; ++r) {
      float sv = s[r] * scale;
      float mx = sv;
      mx = fmaxf(mx, __shfl_xor(mx, 1, 16));
      mx = fmaxf(mx, __shfl_xor(mx, 2, 16));
      mx = fmaxf(mx, __shfl_xor(mx, 4, 16));
      mx = fmaxf(mx, __shfl_xor(mx, 8, 16));
      float mn = fmaxf(mrow[r], mx);
      float al = __expf(mrow[r] - mn);
      mrow[r] = mn;
      float p = __expf(sv - mn);
      s[r] = p;
      float rs = p;
      rs += __shfl_xor(rs, 1, 16);
      rs += __shfl_xor(rs, 2, 16);
      rs += __shfl_xor(rs, 4, 16);
      rs += __shfl_xor(rs, 8, 16);
      lrow[r] = lrow[r] * al + rs;
      alpha[r] = al;
    }
#pragma unroll
    for (int r = 0; r < 8; ++r) {
      acc0[r] *= alpha[r]; acc1[r] *= alpha[r];
      acc2[r] *= alpha[r]; acc3[r] *= alpha[r];
    }

    // ---- P (C layout) -> LDS -> A-layout fragments
    __syncthreads();
#pragma unroll
    for (int r = 0; r < 8; ++r) pt[(r + 8 * half) * 17 + m] = s[r];
    __syncthreads();

    // ---- acc += P * V ; V fragment: B[k][n] = V[key0+k][hd0+n]
    const float* vp = base + (size_t)kt * 16 * 192 + 128;
#pragma unroll
    for (int c = 0; c < 4; ++c) {
      v2f pa;
      pa.x = pt[m * 17 + c * 4 + half * 2 + 0];
      pa.y = pt[m * 17 + c * 4 + half * 2 + 1];
      const int kr = c * 4 + half * 2;
      v2f vf;
      vf.x = vp[(kr + 0) * 192 + 0 + m];
      vf.y = vp[(kr + 1) * 192 + 0 + m];
      acc0 = wmma_f32_16x16x4(pa, vf, acc0);
      vf.x = vp[(kr + 0) * 192 + 16 + m];
      vf.y = vp[(kr + 1) * 192 + 16 + m];
      acc1 = wmma_f32_16x16x4(pa, vf, acc1);
      vf.x = vp[(kr + 0) * 192 + 32 + m];
      vf.y = vp[(kr + 1) * 192 + 32 + m];
      acc2 = wmma_f32_16x16x4(pa, vf, acc2);
      vf.x = vp[(kr + 0) * 192 + 48 + m];
      vf.y = vp[(kr + 1) * 192 + 48 + m];
      acc3 = wmma_f32_16x16x4(pa, vf, acc3);
    }
  }

  // ---- epilogue: aout[b, n, h*64 + col] = acc / l
  float* op = aout + ((size_t)(b * Nk + qt * 16)) * Dk + h * 64;
#pragma unroll
  for (int r = 0; r < 8; ++r) {
    float inv = 1.0f / lrow[r];
    int row = r + 8 * half;
    op[(size_t)row * Dk + 0 + m] = acc0[r] * inv;
    op[(size_t)row * Dk + 16 + m] = acc1[r] * inv;
    op[(size_t)row * Dk + 32 + m] = acc2[r] * inv;
    op[(size_t)row * Dk + 48 + m] = acc3[r] * inv;
  }
}

// ---------------------------------------------------------------------------
// Kernel 3: output projection  out = x @ Wmsa^T + bmsa   (M=8192, N=K=768)
// 32x64 macro-tile per wave, same 2x4 register blocking as kernel 1.
// ---------------------------------------------------------------------------
__global__ __launch_bounds__(128) void proj_kernel(
    const float* __restrict__ x, const float* __restrict__ W,
    const float* __restrict__ bias, float* __restrict__ out) {
  const int wave = threadIdx.x >> 5;
  const int lane = threadIdx.x & 31;
  const int m = lane & 15;
  const int half = lane >> 4;

  int tile = blockIdx.x * 4 + wave;
  const int CB = Dk / 64;  // 12 col blocks of 64
  int cb = tile % CB;
  int rb = tile / CB;  // 256 row blocks of 32

  const float* xa0 = x + (size_t)(rb * 32 + m) * Dk + half * 2;
  const float* xa1 = xa0 + (size_t)16 * Dk;
  const float* wb = W + (size_t)(cb * 64 + m) * Dk + half * 2;

  v8f acc[2][4];
#pragma unroll
  for (int i = 0; i < 2; ++i)
#pragma unroll
    for (int t = 0; t < 4; ++t) acc[i][t] = (v8f){};

  for (int k0 = 0; k0 < Dk; k0 += 64) {
    __builtin_prefetch(xa0 + k0 + 64, 0, 1);
    __builtin_prefetch(xa1 + k0 + 64, 0, 1);
    __builtin_prefetch(wb + k0 + 64, 0, 1);
#pragma unroll
    for (int kk = 0; kk < 64; kk += 4) {
      const int k = k0 + kk;
      v2f a0 = *(const v2f*)(xa0 + k);
      v2f a1 = *(const v2f*)(xa1 + k);
      v2f b0 = *(const v2f*)(wb + k);
      v2f b1 = *(const v2f*)(wb + (size_t)16 * Dk + k);
      v2f b2 = *(const v2f*)(wb + (size_t)32 * Dk + k);
      v2f b3 = *(const v2f*)(wb + (size_t)48 * Dk + k);
      acc[0][0] = wmma_f32_16x16x4(a0, b0, acc[0][0]);
      acc[0][1] = wmma_f32_16x16x4(a0, b1, acc[0][1]);
      acc[0][2] = wmma_f32_16x16x4(a0, b2, acc[0][2]);
      acc[0][3] = wmma_f32_16x16x4(a0, b3, acc[0][3]);
      acc[1][0] = wmma_f32_16x16x4(a1, b0, acc[1][0]);
      acc[1][1] = wmma_f32_16x16x4(a1, b1, acc[1][1]);
      acc[1][2] = wmma_f32_16x16x4(a1, b2, acc[1][2]);
      acc[1][3] = wmma_f32_16x16x4(a1, b3, acc[1][3]);
    }
  }

  float bv[4];
#pragma unroll
  for (int t = 0; t < 4; ++t) bv[t] = bias[cb * 64 + t * 16 + m];

  float* op = out + (size_t)(rb * 32) * Dk + cb * 64 + m;
#pragma unroll
  for (int i = 0; i < 2; ++i)
#pragma unroll
    for (int r = 0; r < 8; ++r) {
      const size_t row = (size_t)(i * 16 + r + 8 * half) * Dk;
#pragma unroll
      for (int t = 0; t < 4; ++t) op[row + t * 16] = acc[i][t][r] + bv[t];
    }
}

// ---------------------------------------------------------------------------
extern "C" void kernel_launch(void* const* d_in, const int* in_sizes, int n_in,
                              void* d_out, int out_size, void* d_ws,
                              size_t ws_size, hipStream_t stream) {
  const float* z    = (const float*)d_in[0];  // [B,N,D]
  const float* Wqkv = (const float*)d_in[1];  // [H,192,D]
  const float* bqkv = (const float*)d_in[2];  // [H,192]
  const float* Wmsa = (const float*)d_in[3];  // [D,D]
  const float* bmsa = (const float*)d_in[4];  // [D]
  float* out = (float*)d_out;                 // [B,N,D]

  float* qkv  = (float*)d_ws;                       // B*H*N*192 floats
  float* aout = qkv + (size_t)Bk * Hk * Nk * 192;   // B*N*D floats

  // 1) QKV projection: 48 * (64 n-blocks) * (3 e-blocks) waves / 4 per block
  qkv_gemm_kernel<<<(Bk * Hk * 64 * 3) / 4, 128, 0, stream>>>(z, Wqkv, bqkv,
                                                              qkv);

  // 2) flash attention: one wave per (b,h,query tile)
  attn_kernel<<<Bk * Hk * (Nk / 16), 32, 0, stream>>>(qkv, aout);

  // 3) output projection: 256 row-blocks * 12 col-blocks waves / 4 per block
  proj_kernel<<<(256 * 12) / 4, 128, 0, stream>>>(aout, Wmsa, bmsa, out);
}